// HeatmapBallDetectionLoss_8778913153573
// MI455X (gfx1250) — compile-verified
//
#include <hip/hip_runtime.h>

typedef float v2f __attribute__((ext_vector_type(2)));
typedef float v4f __attribute__((ext_vector_type(4)));
typedef float v8f __attribute__((ext_vector_type(8)));

// Clamp of -100 in the ln domain == -100/ln2 in the log2 domain (ln2 > 0, monotone).
#define LOG2C (-144.26950408889634f)
#define LN2   (0.6931471805599453f)

// Raw v_log_f32 (log2). p in [0,1) => 1-p is 0 or a normal float, so no denorm
// fixup is needed; log2(0) = -inf is swallowed by the clamp.
__device__ __forceinline__ float clog1m_log2(float p) {
    return fmaxf(__builtin_amdgcn_logf(1.0f - p), LOG2C);
}

// One block per (b,n) row. First BN blocks handle pred_x rows (length W),
// next BN blocks handle pred_y rows (length H). Streaming NT float4 loads,
// log2-domain accumulation, per-wave reduction via V_WMMA_F32_16X16X4_F32
// (all-ones B), ds_swizzle SWAPX16 combine, LDS across waves, thread 0
// applies the one-hot correction and writes row_loss[blk].
__global__ __launch_bounds__(256) void row_bce_kernel(
    const float* __restrict__ pred_x, const float* __restrict__ pred_y,
    const int* __restrict__ tgt, float* __restrict__ row_loss,
    int BN, int W, int H)
{
    const int blk = blockIdx.x;
    const bool isX = blk < BN;
    const int row  = isX ? blk : blk - BN;
    const int L    = isX ? W : H;
    const float* __restrict__ p =
        (isX ? pred_x : pred_y) + (size_t)row * (size_t)L;
    const int tid = threadIdx.x;

    // ---- streaming sum of clamped log2(1-p) over the row ----
    float acc = 0.0f;
    const int L4 = L >> 2;
    const v4f* __restrict__ p4 = (const v4f*)p;
    for (int i = tid; i < L4; i += 256) {
        v4f v = __builtin_nontemporal_load(p4 + i);   // single-pass stream
        acc += clog1m_log2(v.x);
        acc += clog1m_log2(v.y);
        acc += clog1m_log2(v.z);
        acc += clog1m_log2(v.w);
    }
    for (int i = (L4 << 2) + tid; i < L; i += 256)    // tail (none for 1920/1080)
        acc += clog1m_log2(p[i]);

    // ---- wave32 reduction via WMMA: D = A x Ones, A holds lane partials ----
    // Lane l<16: A[l][0]=acc, A[l][1]=0 ; lane l>=16: A[l-16][2]=acc, A[l-16][3]=0.
    // With B = ones, D[m][n] = acc_m + acc_{m+16}. d[0..7] on lanes 0-15 hold
    // rowsums 0-7, on lanes 16-31 rowsums 8-15; sum + xor-16 swap = wave sum.
    v2f a; a.x = acc;  a.y = 0.0f;
    v2f b; b.x = 1.0f; b.y = 1.0f;
    v8f c = {};
    v8f d = __builtin_amdgcn_wmma_f32_16x16x4_f32(
        /*neg_a=*/false, a, /*neg_b=*/false, b,
        /*c_mod=*/(short)0, c, /*reuse_a=*/false, /*reuse_b=*/false);
    float s = d[0] + d[1] + d[2] + d[3] + d[4] + d[5] + d[6] + d[7];
    // SWAPX16: group-of-32 swizzle, xor_mask=0x10, or_mask=0, and_mask=0x1f
    s += __int_as_float(__builtin_amdgcn_ds_swizzle(__float_as_int(s), 0x401F));

    __shared__ float wsum[8];
    if ((tid & 31) == 0) wsum[tid >> 5] = s;
    __syncthreads();

    if (tid == 0) {
        float block_sum = 0.0f;
        #pragma unroll
        for (int w = 0; w < 8; ++w) block_sum += wsum[w];
        const float sum_l1p = LN2 * block_sum;         // back to ln domain

        const int t0 = tgt[2 * row + 0];
        const int t1 = tgt[2 * row + 1];
        const int tx = min(max(t0, 0), W - 1);
        const int ty = min(max(t1, 0), H - 1);
        const float vmask = ((tx == 0) && (ty == 0)) ? 0.0f : 1.0f;

        const int t  = isX ? tx : ty;
        const float pt  = p[t];
        const float lp  = LN2 * fmaxf(__builtin_amdgcn_logf(pt), LOG2C);
        const float l1p = LN2 * fmaxf(__builtin_amdgcn_logf(1.0f - pt), LOG2C);
        // loss = (-lp_t - (sum_l1p - l1p_t)) / L
        const float loss = (-lp + l1p - sum_l1p) / (float)L;
        row_loss[blk] = loss * vmask;
    }
}

// Deterministic final combine: fixed-tree sum of 2*BN row losses and the
// valid-row count (recomputed from targets: BN pairs, trivial bytes).
__global__ __launch_bounds__(1024) void final_reduce_kernel(
    const float* __restrict__ row_loss, const int* __restrict__ tgt,
    float* __restrict__ out, int BN, int W, int H)
{
    __shared__ float sdata[1024];
    const int tid = threadIdx.x;

    float s = 0.0f;
    for (int i = tid; i < 2 * BN; i += 1024) s += row_loss[i];

    float cnt = 0.0f;
    for (int r = tid; r < BN; r += 1024) {
        const int tx = min(max(tgt[2 * r + 0], 0), W - 1);
        const int ty = min(max(tgt[2 * r + 1], 0), H - 1);
        cnt += ((tx == 0) && (ty == 0)) ? 0.0f : 1.0f;
    }

    sdata[tid] = s;
    __syncthreads();
    for (int off = 512; off > 0; off >>= 1) {
        if (tid < off) sdata[tid] += sdata[tid + off];
        __syncthreads();
    }
    const float total = sdata[0];
    __syncthreads();

    sdata[tid] = cnt;
    __syncthreads();
    for (int off = 512; off > 0; off >>= 1) {
        if (tid < off) sdata[tid] += sdata[tid + off];
        __syncthreads();
    }

    if (tid == 0) out[0] = total / fmaxf(sdata[0], 1.0f);
}

extern "C" void kernel_launch(void* const* d_in, const int* in_sizes, int n_in,
                              void* d_out, int out_size, void* d_ws, size_t ws_size,
                              hipStream_t stream) {
    const float* pred_x = (const float*)d_in[0];
    const float* pred_y = (const float*)d_in[1];
    const int*   tgt    = (const int*)d_in[2];
    float*       out    = (float*)d_out;

    const int BN = in_sizes[2] / 2;       // 256*32 = 8192
    const int W  = in_sizes[0] / BN;      // 1920
    const int H  = in_sizes[1] / BN;      // 1080

    float* row_loss = (float*)d_ws;       // 2*BN floats, every slot written each call

    row_bce_kernel<<<2 * BN, 256, 0, stream>>>(pred_x, pred_y, tgt, row_loss, BN, W, H);
    final_reduce_kernel<<<1, 1024, 0, stream>>>(row_loss, tgt, out, BN, W, H);
}